// MTGNNHead_66889820668322
// MI455X (gfx1250) — compile-verified
//
#include <hip/hip_runtime.h>
#include <math.h>

typedef __attribute__((ext_vector_type(2))) float v2f;
typedef __attribute__((ext_vector_type(8))) float v8f;

#define WMMA4(a, b, c) \
  __builtin_amdgcn_wmma_f32_16x16x4_f32(false, (a), false, (b), (short)0, (c), false, false)

// ---------------------------------------------------------------------------
// Generic fp32 WMMA GEMM:  C[M,N] = op( A[M,K](lda) * Bp[K,ldb](padded) + bias )
// Requirements: K % 4 == 0, M % 16 == 0, lda even, ldb % 64 == 0 and B
// zero-padded to ldb columns. No load guards needed; store guards col<N only.
// MODE 0: row-major store (ldc). MODE 1: start-conv scatter (row=(bn*64+l), col=c)
// ---------------------------------------------------------------------------
template <int MODE, bool ACCUM, bool RELUOUT>
__global__ void gemm_wmma_k(const float* __restrict__ A, int lda,
                            const float* __restrict__ B, int ldb,
                            const float* __restrict__ bias,
                            float* __restrict__ C, int ldc,
                            int M, int N, int K) {
  int tilesN = (N + 63) >> 6;
  int wid = blockIdx.x * 8 + (threadIdx.x >> 5);
  if (wid >= (M >> 4) * tilesN) return;             // wave-uniform
  int tm = wid / tilesN, tn = wid % tilesN;
  int lane = threadIdx.x & 31;
  int mrow = lane & 15;                             // A row / B,C col within tile
  int kh = lane >> 4;                               // K-half (0/1)
  const float* Ap = A + (size_t)(tm * 16 + mrow) * lda + 2 * kh;
  const float* Bp = B + (size_t)(2 * kh) * ldb + tn * 64 + mrow;

  v8f acc[4];
#pragma unroll
  for (int j = 0; j < 4; ++j) acc[j] = {0.f, 0.f, 0.f, 0.f, 0.f, 0.f, 0.f, 0.f};

  for (int kk = 0; kk < K; kk += 4) {
    v2f a = *(const v2f*)Ap;                        // 8B aligned: lda even, k even
    Ap += 4;
    v2f bv[4];
#pragma unroll
    for (int j = 0; j < 4; ++j) {
      bv[j].x = Bp[j * 16];
      bv[j].y = Bp[(size_t)ldb + j * 16];
    }
    Bp += (size_t)4 * ldb;
#pragma unroll
    for (int j = 0; j < 4; ++j) acc[j] = WMMA4(a, bv[j], acc[j]);
  }

#pragma unroll
  for (int j = 0; j < 4; ++j) {
    int col = tn * 64 + j * 16 + mrow;
    if (col >= N) continue;
    float bvl = bias ? bias[col] : 0.f;
#pragma unroll
    for (int r = 0; r < 8; ++r) {
      int row = tm * 16 + r + 8 * kh;               // always < M (M%16==0)
      float v = acc[j][r] + bvl;
      if (MODE == 0) {
        size_t idx = (size_t)row * ldc + col;
        if (ACCUM) v += C[idx];
        if (RELUOUT) v = fmaxf(v, 0.f);
        C[idx] = v;
      } else {  // start-conv: row=(bn*64+l), col=c -> C[bn][c][l], stride 64
        int bn = row >> 6, l = row & 63;
        C[((size_t)bn * 32 + col) * 64 + l] = v;
      }
    }
  }
}

template <int MODE, bool ACCUM, bool RELUOUT>
static void launch_gemm(const float* A, int lda, const float* B, int ldb,
                        const float* bias, float* C, int ldc,
                        int M, int N, int K, hipStream_t s) {
  int waves = (M / 16) * ((N + 63) / 64);
  gemm_wmma_k<MODE, ACCUM, RELUOUT><<<(waves + 7) / 8, 256, 0, s>>>(
      A, lda, B, ldb, bias, C, ldc, M, N, K);
}

// ---------------------------------------------------------------------------
// Graph mix-hop propagation step, per batch b (blockIdx.y):
//   O_b[207, 32T] = 0.05 * X_b + 0.95 * (W[208,208] * H_b[207, 32T])
// W is the 208-padded adjacency: column 207 is zero, so the (in-bounds but
// meaningless) H row 207 read contributes exactly zero. Row 207 of W feeds
// only output rows that are never stored. -> zero guards in the K loop.
// ---------------------------------------------------------------------------
__global__ void prop_wmma_k(const float* __restrict__ W,
                            const float* __restrict__ H,
                            const float* __restrict__ X,
                            float* __restrict__ O, int T) {
  int Ncol = 32 * T;
  int tilesN = Ncol >> 6;
  int wid = blockIdx.x * 8 + (threadIdx.x >> 5);
  if (wid >= 13 * tilesN) return;                   // tilesM = ceil(207/16) = 13
  int b = blockIdx.y;
  size_t base = (size_t)b * 207 * Ncol;
  int tm = wid / tilesN, tn = wid % tilesN;
  int lane = threadIdx.x & 31, mrow = lane & 15, kh = lane >> 4;
  int arow = tm * 16 + mrow;                        // <= 207, in padded bounds
  const float* Wp = W + (size_t)arow * 208 + 2 * kh;
  const float* Hp = H + base + (size_t)(2 * kh) * Ncol + tn * 64 + mrow;

  v8f acc[4];
#pragma unroll
  for (int j = 0; j < 4; ++j) acc[j] = {0.f, 0.f, 0.f, 0.f, 0.f, 0.f, 0.f, 0.f};

  for (int kk = 0; kk < 208; kk += 4) {
    v2f a = *(const v2f*)Wp;                        // 8B aligned (208 even)
    Wp += 4;
    v2f bv[4];
#pragma unroll
    for (int j = 0; j < 4; ++j) {
      bv[j].x = Hp[j * 16];
      bv[j].y = Hp[(size_t)Ncol + j * 16];
    }
    Hp += (size_t)4 * Ncol;
#pragma unroll
    for (int j = 0; j < 4; ++j) acc[j] = WMMA4(a, bv[j], acc[j]);
  }

#pragma unroll
  for (int j = 0; j < 4; ++j) {
    int col = tn * 64 + j * 16 + mrow;
#pragma unroll
    for (int r = 0; r < 8; ++r) {
      int row = tm * 16 + r + 8 * kh;
      if (row < 207) {
        size_t idx = base + (size_t)row * Ncol + col;
        O[idx] = 0.05f * X[idx] + 0.95f * acc[j][r];
      }
    }
  }
}

// ---------------------------------------------------------------------------
// Fused 1x1 mix conv (per bn): Out[32,T] = bias + Wm[32,96] * [g; p1; p2]
// PASS2 additionally adds the pass-1 partial (macc) + residual, then does
// channel LayerNorm and writes in-place over g.
// ---------------------------------------------------------------------------
template <bool PASS2>
__global__ void mix_wmma_k(const float* __restrict__ g,
                           const float* __restrict__ p1,
                           const float* __restrict__ p2,
                           const float* __restrict__ Wm,
                           const float* __restrict__ bias,
                           float* __restrict__ macc,
                           float* __restrict__ gout,
                           const float* __restrict__ hin, int Tin,
                           const float* __restrict__ gamma,
                           const float* __restrict__ beta, int T) {
  __shared__ float catB[96 * 64];
  __shared__ float outT[32 * 64];
  int bn = blockIdx.x, tid = threadIdx.x;
  size_t gbase = (size_t)bn * 32 * T;

  for (int idx = tid; idx < 96 * 64; idx += 256) {   // stage B into LDS
    int r = idx >> 6, c = idx & 63;
    float v = 0.f;
    if (c < T) {
      const float* src = (r < 32) ? g : (r < 64) ? p1 : p2;
      v = src[gbase + (size_t)(r & 31) * T + c];
    }
    catB[idx] = v;
  }
  __syncthreads();

  int wid = tid >> 5, lane = tid & 31, mrow = lane & 15, kh = lane >> 4;
  int tm = wid >> 2, tn = wid & 3;                  // 2 x 4 tiles = 8 waves
  const float* Wp = Wm + (size_t)(tm * 16 + mrow) * 96 + 2 * kh;
  const float* Cb = catB + (size_t)(2 * kh) * 64 + tn * 16 + mrow;
  v8f acc = {0.f, 0.f, 0.f, 0.f, 0.f, 0.f, 0.f, 0.f};
  for (int kk = 0; kk < 96; kk += 4) {
    v2f a = *(const v2f*)Wp;                        // 8B aligned (96 even)
    Wp += 4;
    v2f b;
    b.x = Cb[0];
    b.y = Cb[64];
    Cb += 4 * 64;
    acc = WMMA4(a, b, acc);
  }
#pragma unroll
  for (int r = 0; r < 8; ++r) {
    int row = tm * 16 + r + 8 * kh;
    outT[row * 64 + tn * 16 + mrow] = acc[r] + bias[row];
  }
  __syncthreads();

  if (!PASS2) {
    for (int idx = tid; idx < 32 * T; idx += 256) {
      int c = idx / T, t = idx % T;
      macc[gbase + idx] = outT[c * 64 + t];
    }
  } else if (tid < T) {
    int t = tid;
    float vals[32], s = 0.f, sq = 0.f;
#pragma unroll
    for (int c = 0; c < 32; ++c) {
      float v = outT[c * 64 + t] + macc[gbase + (size_t)c * T + t] +
                hin[(size_t)bn * 32 * Tin + (size_t)c * Tin + (Tin - T) + t];
      vals[c] = v; s += v; sq += v * v;
    }
    float mu = s * (1.f / 32.f);
    float var = sq * (1.f / 32.f) - mu * mu;
    float rs = rsqrtf(var + 1e-5f);
#pragma unroll
    for (int c = 0; c < 32; ++c)
      gout[gbase + (size_t)c * T + t] = (vals[c] - mu) * rs * gamma[c] + beta[c];
  }
}

// ---------------------------------------------------------------------------
// Inception temporal conv (k=2,3,6,7, 8 out-ch each) + GLU. One block per bn.
// ---------------------------------------------------------------------------
__global__ void inception_glu_k(const float* __restrict__ hin, int Tin,
                                float* __restrict__ gout, int Tout,
                                const float* fw0, const float* fw1,
                                const float* fw2, const float* fw3,
                                const float* fb0, const float* fb1,
                                const float* fb2, const float* fb3,
                                const float* gw0, const float* gw1,
                                const float* gw2, const float* gw3,
                                const float* gb0, const float* gb1,
                                const float* gb2, const float* gb3) {
  __shared__ float sh[32 * 64];
  __shared__ float sFW[4608], sGW[4608];
  __shared__ float sFB[32], sGB[32];
  int bn = blockIdx.x, tid = threadIdx.x;
  for (int i = tid; i < 32 * Tin; i += 256) sh[i] = hin[(size_t)bn * 32 * Tin + i];
  const float* fwp[4] = {fw0, fw1, fw2, fw3};
  const float* gwp[4] = {gw0, gw1, gw2, gw3};
  const float* fbp[4] = {fb0, fb1, fb2, fb3};
  const float* gbp[4] = {gb0, gb1, gb2, gb3};
  const int ks[4] = {2, 3, 6, 7};
  const int wof[4] = {0, 512, 1280, 2816};
#pragma unroll
  for (int gi = 0; gi < 4; ++gi) {
    int n = 8 * 32 * ks[gi];
    for (int i = tid; i < n; i += 256) {
      sFW[wof[gi] + i] = fwp[gi][i];
      sGW[wof[gi] + i] = gwp[gi][i];
    }
  }
  if (tid < 32) { sFB[tid] = fbp[tid >> 3][tid & 7]; sGB[tid] = gbp[tid >> 3][tid & 7]; }
  __syncthreads();

  for (int o = tid; o < 32 * Tout; o += 256) {
    int co = o / Tout, t = o % Tout;
    int gi = co >> 3, k = ks[gi];
    int wbase = wof[gi] + (co & 7) * 32 * k;
    int t0 = 7 - k + t;                              // align truncated outputs
    float af = sFB[co], ag = sGB[co];
    for (int ci = 0; ci < 32; ++ci) {
      const float* hr = sh + ci * Tin + t0;
      const float* wf = sFW + wbase + ci * k;
      const float* wg = sGW + wbase + ci * k;
      for (int j = 0; j < k; ++j) { af += hr[j] * wf[j]; ag += hr[j] * wg[j]; }
    }
    gout[(size_t)bn * 32 * Tout + o] = tanhf(af) * (1.0f / (1.0f + expf(-ag)));
  }
}

// ---------------------------------------------------------------------------
// Adjacency: v1/v2 = tanh(3*(emb@W^T+b)); adj = relu(tanh(3*(v1 v2^T - v2 v1^T)))
// masked to top-20 per row (stable tie-break, matching lax.top_k).
// Output stride 208; column 207 zeroed (K padding for the prop GEMM).
// ---------------------------------------------------------------------------
__global__ void adj_embed_k(const float* __restrict__ emb,
                            const float* __restrict__ w1, const float* __restrict__ b1,
                            const float* __restrict__ w2, const float* __restrict__ b2,
                            float* __restrict__ v1, float* __restrict__ v2) {
  int n = blockIdx.x, o = threadIdx.x;
  if (o >= 40) return;
  float s1 = b1[o], s2 = b2[o];
  for (int d = 0; d < 40; ++d) {
    float e = emb[n * 40 + d];
    s1 += e * w1[o * 40 + d];
    s2 += e * w2[o * 40 + d];
  }
  v1[n * 40 + o] = tanhf(3.0f * s1);
  v2[n * 40 + o] = tanhf(3.0f * s2);
}

__global__ void adj_topk_k(const float* __restrict__ v1, const float* __restrict__ v2,
                           float* __restrict__ adj, float* __restrict__ adjT) {
  __shared__ float av[224], aor[224], msk[224];
  __shared__ float rv[256];
  __shared__ int ri[256];
  int n = blockIdx.x, tid = threadIdx.x;
  if (tid < 224) {
    float val = -1.f;
    if (tid < 207) {
      float s = 0.f;
      for (int d = 0; d < 40; ++d)
        s += v1[n * 40 + d] * v2[tid * 40 + d] - v2[n * 40 + d] * v1[tid * 40 + d];
      val = fmaxf(tanhf(3.0f * s), 0.f);
    }
    av[tid] = val; aor[tid] = val; msk[tid] = 0.f;
  }
  __syncthreads();
  for (int it = 0; it < 20; ++it) {
    rv[tid] = (tid < 207) ? av[tid] : -2.f;
    ri[tid] = tid;
    __syncthreads();
    for (int s = 128; s > 0; s >>= 1) {
      if (tid < s) {
        float vo = rv[tid + s]; int io = ri[tid + s];
        if (vo > rv[tid] || (vo == rv[tid] && io < ri[tid])) { rv[tid] = vo; ri[tid] = io; }
      }
      __syncthreads();
    }
    if (tid == 0) { msk[ri[0]] = 1.f; av[ri[0]] = -2.f; }
    __syncthreads();
  }
  if (tid < 207) {
    float o = (msk[tid] > 0.f) ? aor[tid] : 0.f;
    adj[n * 208 + tid] = o;
    adjT[tid * 208 + n] = o;
  }
  if (tid == 0) {                        // zero the K-padding column
    adj[n * 208 + 207] = 0.f;
    adjT[n * 208 + 207] = 0.f;
  }
}

// ---------------------------------------------------------------------------
// Weight repacks and misc elementwise.
// ---------------------------------------------------------------------------
__global__ void pack_skip0_k(const float* __restrict__ w, float* __restrict__ bp) {
  int i = blockIdx.x * 256 + threadIdx.x;            // over 8192*64
  if (i >= 8192 * 64) return;
  int k = i >> 6, s = i & 63;                        // k = l*128 + d
  bp[i] = w[s * 8192 + (k & 127) * 64 + (k >> 7)];
}

__global__ void pack_T_k(const float* __restrict__ w, float* __restrict__ bp,
                         int K, int N, int Npad) {
  int i = blockIdx.x * 256 + threadIdx.x;
  if (i >= K * Npad) return;
  int k = i / Npad, n = i % Npad;
  bp[i] = (n < N) ? w[n * K + k] : 0.f;
}

__global__ void relu_kk(const float* __restrict__ x, float* __restrict__ y, int n) {
  int i = blockIdx.x * 256 + threadIdx.x;
  if (i < n) y[i] = fmaxf(x[i], 0.f);
}

// ---------------------------------------------------------------------------
extern "C" void kernel_launch(void* const* d_in, const int* in_sizes, int n_in,
                              void* d_out, int out_size, void* d_ws, size_t ws_size,
                              hipStream_t stream) {
  (void)in_sizes; (void)n_in; (void)out_size; (void)ws_size;
  const float* x      = (const float*)d_in[0];       // [6624,64,128]
  const float* startW = (const float*)d_in[1];
  const float* startB = (const float*)d_in[2];
  const float* skip0W = (const float*)d_in[3];
  const float* skip0B = (const float*)d_in[4];
  const float* emb    = (const float*)d_in[5];
  const float* lin1W  = (const float*)d_in[6];
  const float* lin1B  = (const float*)d_in[7];
  const float* lin2W  = (const float*)d_in[8];
  const float* lin2B  = (const float*)d_in[9];
  // filter: 10..33 (3 layers x 4 kernels x (w,b)); gate: 34..57
  // mixf: 58..63; mixb: 64..69; skip: 70..75; ln: 76..81; skipE: 82,83
  const float* end1W = (const float*)d_in[84];
  const float* end1B = (const float*)d_in[85];
  const float* end2W = (const float*)d_in[86];
  const float* end2B = (const float*)d_in[87];

  float* ws = (float*)d_ws;
  size_t off = 0;
  auto alloc = [&](size_t n) { float* p = ws + off; off += (n + 63) & ~(size_t)63; return p; };

  float* adj    = alloc(208 * 208);     // padded: col 207 zero, row 207 unused
  float* adjT   = alloc(208 * 208);
  float* v1     = alloc(207 * 40);
  float* v2     = alloc(207 * 40);
  float* pSkip0 = alloc(8192 * 64);
  float* pStart = alloc(128 * 64);      // padded to 64 cols
  float* pEnd1  = alloc(64 * 128);
  float* pEnd2  = alloc(128 * 64);      // padded to 64 cols
  float* pSkipL[3]; int TT[3] = {58, 52, 46};
  for (int l = 0; l < 3; ++l) pSkipL[l] = alloc((size_t)32 * TT[l] * 64);
  float* pSkipE  = alloc((size_t)32 * 46 * 64);
  float* skipAcc = alloc((size_t)6624 * 64);
  float* sE      = alloc((size_t)6624 * 64);
  float* Ebuf    = alloc((size_t)6624 * 128);
  const size_t SZH = (size_t)6624 * 32 * 64;
  float* hA   = alloc(SZH);
  float* hB   = alloc(SZH);
  float* f1   = alloc(SZH);
  float* f2   = alloc(SZH);
  float* macc = alloc(SZH);

  // ---- weight repacks (deterministic each call) ----
  pack_skip0_k<<<(8192 * 64 + 255) / 256, 256, 0, stream>>>(skip0W, pSkip0);
  pack_T_k<<<(128 * 64 + 255) / 256, 256, 0, stream>>>(startW, pStart, 128, 32, 64);
  pack_T_k<<<(64 * 128 + 255) / 256, 256, 0, stream>>>(end1W, pEnd1, 64, 128, 128);
  pack_T_k<<<(128 * 64 + 255) / 256, 256, 0, stream>>>(end2W, pEnd2, 128, 12, 64);
  for (int l = 0; l < 3; ++l) {
    int K = 32 * TT[l];
    pack_T_k<<<(K * 64 + 255) / 256, 256, 0, stream>>>((const float*)d_in[70 + 2 * l],
                                                       pSkipL[l], K, 64, 64);
  }
  pack_T_k<<<(32 * 46 * 64 + 255) / 256, 256, 0, stream>>>((const float*)d_in[82],
                                                           pSkipE, 32 * 46, 64, 64);

  // ---- adjacency ----
  adj_embed_k<<<207, 64, 0, stream>>>(emb, lin1W, lin1B, lin2W, lin2B, v1, v2);
  adj_topk_k<<<207, 256, 0, stream>>>(v1, v2, adj, adjT);

  // ---- skip0 (full-length conv == GEMM [6624,8192]x[8192,64]) ----
  launch_gemm<0, false, false>(x, 8192, pSkip0, 64, skip0B, skipAcc, 64,
                               6624, 64, 8192, stream);
  // ---- start 1x1 conv: [423936,128]x[128,32], scattered to [bn,c,l] ----
  launch_gemm<1, false, false>(x, 128, pStart, 64, startB, hA, 0,
                               423936, 32, 128, stream);

  float* hIn = hA;
  float* g = hB;
  int Tin = 64;
  for (int l = 0; l < 3; ++l) {
    int T = Tin - 6;
    int fb = 10 + 8 * l, gb = 34 + 8 * l;
    inception_glu_k<<<6624, 256, 0, stream>>>(
        hIn, Tin, g, T,
        (const float*)d_in[fb + 0], (const float*)d_in[fb + 2],
        (const float*)d_in[fb + 4], (const float*)d_in[fb + 6],
        (const float*)d_in[fb + 1], (const float*)d_in[fb + 3],
        (const float*)d_in[fb + 5], (const float*)d_in[fb + 7],
        (const float*)d_in[gb + 0], (const float*)d_in[gb + 2],
        (const float*)d_in[gb + 4], (const float*)d_in[gb + 6],
        (const float*)d_in[gb + 1], (const float*)d_in[gb + 3],
        (const float*)d_in[gb + 5], (const float*)d_in[gb + 7]);

    // skip conv: GEMM [6624, 32T] x [32T, 64], accumulate (+bias)
    launch_gemm<0, true, false>(g, 32 * T, pSkipL[l], 64,
                                (const float*)d_in[71 + 2 * l],
                                skipAcc, 64, 6624, 64, 32 * T, stream);

    int tilesN = (32 * T) >> 6;
    dim3 pgrid((13 * tilesN + 7) / 8, 32);
    // forward direction: out = adjT * h
    prop_wmma_k<<<pgrid, 256, 0, stream>>>(adjT, g, g, f1, T);
    prop_wmma_k<<<pgrid, 256, 0, stream>>>(adjT, f1, g, f2, T);
    mix_wmma_k<false><<<6624, 256, 0, stream>>>(
        g, f1, f2, (const float*)d_in[58 + 2 * l], (const float*)d_in[59 + 2 * l],
        macc, nullptr, nullptr, 0, nullptr, nullptr, T);
    // backward direction: out = adj * h (einsum with A^T)
    prop_wmma_k<<<pgrid, 256, 0, stream>>>(adj, g, g, f1, T);
    prop_wmma_k<<<pgrid, 256, 0, stream>>>(adj, f1, g, f2, T);
    mix_wmma_k<true><<<6624, 256, 0, stream>>>(
        g, f1, f2, (const float*)d_in[64 + 2 * l], (const float*)d_in[65 + 2 * l],
        macc, g, hIn, Tin,
        (const float*)d_in[76 + 2 * l], (const float*)d_in[77 + 2 * l], T);

    float* tmp = hIn; hIn = g; g = tmp;  // LN result (in g buffer) becomes input
    Tin = T;
  }

  // ---- skipE accumulate, head ----
  launch_gemm<0, true, false>(hIn, 32 * 46, pSkipE, 64, (const float*)d_in[83],
                              skipAcc, 64, 6624, 64, 32 * 46, stream);
  relu_kk<<<(6624 * 64 + 255) / 256, 256, 0, stream>>>(skipAcc, sE, 6624 * 64);
  launch_gemm<0, false, true>(sE, 64, pEnd1, 128, end1B, Ebuf, 128,
                              6624, 128, 64, stream);
  launch_gemm<0, false, false>(Ebuf, 128, pEnd2, 64, end2B, (float*)d_out, 12,
                               6624, 12, 128, stream);
}